// HungarianMatcher_34969623724727
// MI455X (gfx1250) — compile-verified
//
#include <hip/hip_runtime.h>
#include <cstdint>
#include <cstddef>

// Problem constants (match reference).
static constexpr int NB = 128;   // batch
static constexpr int NQ = 1000;  // queries
static constexpr int NC = 256;   // classes
static constexpr int NT = 300;   // targets
static constexpr int QTILE = 8;  // one q per wave, 8 waves (256 threads) per block
static constexpr int TGT_PAD = 512; // padded LDS target count (keeps all lanes active on async loads)

__global__ __launch_bounds__(256)
void HungarianMatcher_cost_kernel(const float* __restrict__ logits,   // [B,Q,C]
                                  const float* __restrict__ pboxes,   // [B,Q,4]
                                  const int*   __restrict__ labels,   // [B,T]
                                  const float* __restrict__ tboxes,   // [B,T,4]
                                  float* __restrict__ out)            // [B,Q,T]
{
    __shared__ int   s_lab[TGT_PAD];        // 2 KB
    __shared__ float s_tb [TGT_PAD * 4];    // 8 KB
    __shared__ float s_negp[QTILE * NC];    // 8 KB : -softmax prob per (wave, class)

    const int tid  = threadIdx.x;
    const int wave = tid >> 5;
    const int lane = tid & 31;
    const int blk  = blockIdx.x;
    const int b    = blk / (NQ / QTILE);
    const int q0   = (blk % (NQ / QTILE)) * QTILE;
    const int q    = q0 + wave;

    // ------------------------------------------------------------------
    // Stage 0: asynchronously stage this batch's targets into LDS
    // (gfx1250 GLOBAL_LOAD_ASYNC_TO_LDS_*, tracked by ASYNCcnt).
    // All 32 lanes of every wave stay active; extra lanes load a clamped
    // (duplicate) element into LDS padding slots that are never read.
    // ------------------------------------------------------------------
    {
        const int*   glab = labels + (size_t)b * NT;
        const float* gtb  = tboxes + (size_t)b * NT * 4;
#pragma unroll
        for (int r = 0; r < 2; ++r) {
            const int i  = tid + r * 256;          // 0..511
            const int gi = (i < NT) ? i : (NT - 1);
            unsigned lds_off = (unsigned)(uintptr_t)(&s_lab[i]);
            unsigned long long ga = (unsigned long long)(uintptr_t)(glab + gi);
            asm volatile("global_load_async_to_lds_b32 %0, %1, off"
                         :: "v"(lds_off), "v"(ga) : "memory");
        }
#pragma unroll
        for (int r = 0; r < 2; ++r) {
            const int i  = tid + r * 256;
            const int gi = (i < NT) ? i : (NT - 1);
            unsigned lds_off = (unsigned)(uintptr_t)(&s_tb[i * 4]);
            unsigned long long ga = (unsigned long long)(uintptr_t)(gtb + (size_t)gi * 4);
            asm volatile("global_load_async_to_lds_b128 %0, %1, off"
                         :: "v"(lds_off), "v"(ga) : "memory");
        }
    }

    // ------------------------------------------------------------------
    // Stage 1 (overlapped with async staging): softmax of this wave's
    // q-row.  Lane handles 8 contiguous classes (2x global_load_b128).
    // Wave32 butterfly reductions for max and sum.
    // ------------------------------------------------------------------
    const float* row = logits + ((size_t)b * NQ + q) * NC;
    float4 x0 = ((const float4*)row)[lane * 2 + 0];
    float4 x1 = ((const float4*)row)[lane * 2 + 1];

    float m = fmaxf(fmaxf(fmaxf(x0.x, x0.y), fmaxf(x0.z, x0.w)),
                    fmaxf(fmaxf(x1.x, x1.y), fmaxf(x1.z, x1.w)));
#pragma unroll
    for (int o = 16; o > 0; o >>= 1)
        m = fmaxf(m, __shfl_xor(m, o, 32));

    float4 e0, e1;
    e0.x = __expf(x0.x - m); e0.y = __expf(x0.y - m);
    e0.z = __expf(x0.z - m); e0.w = __expf(x0.w - m);
    e1.x = __expf(x1.x - m); e1.y = __expf(x1.y - m);
    e1.z = __expf(x1.z - m); e1.w = __expf(x1.w - m);

    float s = (e0.x + e0.y + e0.z + e0.w) + (e1.x + e1.y + e1.z + e1.w);
#pragma unroll
    for (int o = 16; o > 0; o >>= 1)
        s += __shfl_xor(s, o, 32);

    const float ninv = -1.0f / s;   // store negated prob: cost_class = -p
    float4 p0 = make_float4(e0.x * ninv, e0.y * ninv, e0.z * ninv, e0.w * ninv);
    float4 p1 = make_float4(e1.x * ninv, e1.y * ninv, e1.z * ninv, e1.w * ninv);
    ((float4*)&s_negp[wave * NC + lane * 8])[0] = p0;
    ((float4*)&s_negp[wave * NC + lane * 8])[1] = p1;

    // This wave's predicted box (broadcast load, L2/L0 hot).
    const float4 pb = *(const float4*)(pboxes + ((size_t)b * NQ + q) * 4);

    // Wait for this wave's async LDS staging, then workgroup barrier so
    // every wave sees the full target tables + all prob rows.
    asm volatile("s_wait_asynccnt 0" ::: "memory");
    __syncthreads();

    // ------------------------------------------------------------------
    // Stage 2: emit C[b,q,t] = L1(pred_box, tgt_box[t]) - p[label[t]].
    // Each lane produces 4 consecutive t's -> one global_store_b128.
    // 300 = 75 groups of 4; row stride 1200 B keeps float4 alignment.
    // ------------------------------------------------------------------
    const float* negp_row = &s_negp[wave * NC];
    float4* out_row = (float4*)(out + ((size_t)b * NQ + q) * NT);

    for (int g = lane; g < NT / 4; g += 32) {
        const int t = g * 4;
        float res[4];
#pragma unroll
        for (int j = 0; j < 4; ++j) {
            const int   lab = s_lab[t + j];
            const float cls = negp_row[lab];
            const float4 tb = ((const float4*)s_tb)[t + j];
            const float cb  = fabsf(pb.x - tb.x) + fabsf(pb.y - tb.y)
                            + fabsf(pb.z - tb.z) + fabsf(pb.w - tb.w);
            res[j] = cb + cls;   // COST_BBOX = COST_CLASS = 1.0
        }
        out_row[g] = make_float4(res[0], res[1], res[2], res[3]);
    }
}

extern "C" void kernel_launch(void* const* d_in, const int* in_sizes, int n_in,
                              void* d_out, int out_size, void* d_ws, size_t ws_size,
                              hipStream_t stream) {
    const float* logits = (const float*)d_in[0];  // pred_logits [B,Q,C] f32
    const float* pboxes = (const float*)d_in[1];  // pred_boxes  [B,Q,4] f32
    const int*   labels = (const int*)  d_in[2];  // tgt_labels  [B,T]   i32
    const float* tboxes = (const float*)d_in[3];  // tgt_boxes   [B,T,4] f32
    float* out = (float*)d_out;                   // C_mat [B,Q,T] f32

    const int blocks = NB * (NQ / QTILE);         // 16000
    HungarianMatcher_cost_kernel<<<blocks, 256, 0, stream>>>(
        logits, pboxes, labels, tboxes, out);
}